// FastSpeech2Loss_78262894067871
// MI455X (gfx1250) — compile-verified
//
#include <hip/hip_runtime.h>
#include <hip/hip_bf16.h>
#include <math.h>

// Problem constants (fixed by the reference harness)
constexpr int BATCH   = 64;
constexpr int T_TEXT  = 200;
constexpr int T_FEATS = 2600;
constexpr int ODIM    = 80;

constexpr int TILES_PER_B = (T_FEATS + 15) / 16;   // 163 (last tile partial: 8 frames)
constexpr int TOTAL_TILES = BATCH * TILES_PER_B;   // 10432
constexpr int NBLK_A      = 512;                   // fixed grid -> deterministic partials

typedef __attribute__((ext_vector_type(2))) float v2f;
typedef __attribute__((ext_vector_type(8))) float v8f;

// ---------------------------------------------------------------------------
// Kernel A: single streaming pass over the three (B, T_feats, odim) tensors.
//  - masked L1 sum: |before-ys| + |after-ys| for frames f < olens[b]
//  - per-frame row sums of ys and after via V_WMMA_F32_16X16X4_F32 with a
//    ones B-matrix (D[m,n] = sum_k A[m,k]).
//
// b128 streaming: per 8-dim chunk, lane l<16 loads float4 dims {k0..k0+3} of
// frame m=l&15, lane l>=16 loads dims {k0+4..k0+7} of the same frame. With a
// ones B-matrix the K-slot assignment is irrelevant for a row sum, so two
// WMMAs ({x,y} then {z,w}) fold all 8 dims into row m exactly once.
// C/D layout: lane 0 holds rows 0..7, lane 16 holds rows 8..15 (N=0 replica).
// ---------------------------------------------------------------------------
__global__ void __launch_bounds__(256)
fs2_stream_kernel(const float* __restrict__ after,
                  const float* __restrict__ before,
                  const float* __restrict__ ys,
                  const int*   __restrict__ olens,
                  float* __restrict__ fsum_ys,
                  float* __restrict__ fsum_af,
                  float* __restrict__ l1_part)
{
    __shared__ float warpsum[8];
    const int tid   = threadIdx.x;
    const int lane  = tid & 31;
    const int wid   = tid >> 5;
    const int gwave = blockIdx.x * 8 + wid;
    const int nwave = gridDim.x * 8;

    const v2f ones = {1.0f, 1.0f};
    float l1 = 0.0f;

    for (int tile = gwave; tile < TOTAL_TILES; tile += nwave) {
        const int b  = tile / TILES_PER_B;
        const int f0 = (tile % TILES_PER_B) * 16;
        const int f  = f0 + (lane & 15);
        const int fc = (f < T_FEATS) ? f : (T_FEATS - 1);   // clamp partial tile
        const int olen = olens[b];
        const float lm = (f < olen) ? 1.0f : 0.0f;          // out_mask (olen<=2400<2600)

        // lane<16 covers dims [k0, k0+4), lane>=16 covers [k0+4, k0+8)
        const size_t rowbase = ((size_t)b * T_FEATS + fc) * ODIM + ((lane >> 4) << 2);

        v8f accY = {0.f,0.f,0.f,0.f,0.f,0.f,0.f,0.f};
        v8f accA = {0.f,0.f,0.f,0.f,0.f,0.f,0.f,0.f};

        #pragma unroll 2
        for (int k0 = 0; k0 < ODIM; k0 += 8) {
            const size_t idx = rowbase + (size_t)k0;
            const float4 y  = *(const float4*)(ys     + idx);
            const float4 a  = *(const float4*)(after  + idx);
            const float4 bf = *(const float4*)(before + idx);

            l1 += lm * (fabsf(bf.x - y.x) + fabsf(bf.y - y.y) +
                        fabsf(bf.z - y.z) + fabsf(bf.w - y.w) +
                        fabsf(a.x  - y.x) + fabsf(a.y  - y.y) +
                        fabsf(a.z  - y.z) + fabsf(a.w  - y.w));

            v2f ay0 = {y.x, y.y}, ay1 = {y.z, y.w};
            v2f aa0 = {a.x, a.y}, aa1 = {a.z, a.w};
            accY = __builtin_amdgcn_wmma_f32_16x16x4_f32(
                       false, ay0, false, ones, (short)0, accY, false, false);
            accY = __builtin_amdgcn_wmma_f32_16x16x4_f32(
                       false, ay1, false, ones, (short)0, accY, false, false);
            accA = __builtin_amdgcn_wmma_f32_16x16x4_f32(
                       false, aa0, false, ones, (short)0, accA, false, false);
            accA = __builtin_amdgcn_wmma_f32_16x16x4_f32(
                       false, aa1, false, ones, (short)0, accA, false, false);
        }

        // Extract N=0 replica of the row sums: lane 0 -> M=0..7, lane 16 -> M=8..15
        if ((lane & 15) == 0) {
            const int mb = (lane < 16) ? 0 : 8;
            const size_t ob = (size_t)b * T_FEATS;
            #pragma unroll
            for (int r = 0; r < 8; ++r) {
                const int fr = f0 + mb + r;
                if (fr < T_FEATS) {
                    fsum_ys[ob + fr] = accY[r];
                    fsum_af[ob + fr] = accA[r];
                }
            }
        }
    }

    // Deterministic wave32 + block reduction of the L1 partial
    for (int off = 16; off > 0; off >>= 1)
        l1 += __shfl_xor(l1, off, 32);
    if (lane == 0) warpsum[wid] = l1;
    __syncthreads();
    if (tid == 0) {
        float s = 0.0f;
        for (int w = 0; w < 8; ++w) s += warpsum[w];
        l1_part[blockIdx.x] = s;
    }
}

// ---------------------------------------------------------------------------
// Kernel B: per-batch token losses. Replaces searchsorted/segment_sum with an
// exclusive cumsum of ds: token t owns frames [excl[t], excl[t]+ds[t]).
// ---------------------------------------------------------------------------
__global__ void __launch_bounds__(256)
fs2_token_kernel(const float* __restrict__ d_outs,
                 const float* __restrict__ e_outs,
                 const float* __restrict__ es,
                 const int*   __restrict__ ds,
                 const int*   __restrict__ ilens,
                 const float* __restrict__ fsum_ys,
                 const float* __restrict__ fsum_af,
                 float* __restrict__ dur_part,
                 float* __restrict__ en_part,
                 float* __restrict__ mean_part)
{
    const int b   = blockIdx.x;
    const int tid = threadIdx.x;

    __shared__ int   excl[T_TEXT];
    __shared__ float rd[256], re[256], rm[256];

    if (tid == 0) {
        int c = 0;
        for (int t = 0; t < T_TEXT; ++t) { excl[t] = c; c += ds[b * T_TEXT + t]; }
    }
    __syncthreads();

    float dv = 0.0f, ev = 0.0f, mv = 0.0f;
    const int ilen = ilens[b];
    if (tid < T_TEXT && tid < ilen) {
        const int t = tid;
        const int c = ds[b * T_TEXT + t];

        const float ld = logf((float)c + 1.0f);
        const float xd = d_outs[b * T_TEXT + t] - ld;
        dv = xd * xd;

        const float xe = e_outs[b * T_TEXT + t] - es[b * T_TEXT + t];
        ev = xe * xe;

        if (c > 0) {
            const size_t base = (size_t)b * T_FEATS + excl[t];
            float sy = 0.0f, sa = 0.0f;
            for (int i = 0; i < c; ++i) {           // c <= 12
                sy += fsum_ys[base + i];
                sa += fsum_af[base + i];
            }
            const float dm = (sa - sy) / ((float)c * (float)ODIM);  // m_outs - ms
            mv = dm * dm;
        }
    }

    rd[tid] = dv; re[tid] = ev; rm[tid] = mv;
    __syncthreads();
    for (int s = 128; s > 0; s >>= 1) {
        if (tid < s) {
            rd[tid] += rd[tid + s];
            re[tid] += re[tid + s];
            rm[tid] += rm[tid + s];
        }
        __syncthreads();
    }
    if (tid == 0) {
        dur_part[b]  = rd[0];
        en_part[b]   = re[0];
        mean_part[b] = rm[0];
    }
}

// ---------------------------------------------------------------------------
// Kernel C: finalize (single thread, fixed order -> deterministic).
// Output order: l1, duration, energy, pitch_max, mean.
// ---------------------------------------------------------------------------
__global__ void fs2_final_kernel(const float* __restrict__ l1_part,
                                 const float* __restrict__ dur_part,
                                 const float* __restrict__ en_part,
                                 const float* __restrict__ mean_part,
                                 const int*   __restrict__ ilens,
                                 const int*   __restrict__ olens,
                                 const float* __restrict__ p_m_outs,
                                 const float* __restrict__ pms,
                                 float* __restrict__ out)
{
    if (threadIdx.x != 0 || blockIdx.x != 0) return;

    float l1 = 0.0f;
    for (int i = 0; i < NBLK_A; ++i) l1 += l1_part[i];

    float du = 0.0f, en = 0.0f, mn = 0.0f, pm = 0.0f;
    long  si = 0, so = 0;
    for (int i = 0; i < BATCH; ++i) {
        du += dur_part[i];
        en += en_part[i];
        mn += mean_part[i];
        si += ilens[i];
        so += olens[i];
        const float d = p_m_outs[i] - pms[i];
        pm += d * d;
    }

    out[0] = l1 / ((float)so * (float)ODIM);  // l1_loss
    out[1] = du / (float)si;                  // duration_loss
    out[2] = en / (float)si;                  // energy_loss
    out[3] = pm / (float)BATCH;               // pitch_max_loss
    out[4] = mn / (float)si;                  // mean_loss
}

// ---------------------------------------------------------------------------
extern "C" void kernel_launch(void* const* d_in, const int* in_sizes, int n_in,
                              void* d_out, int out_size, void* d_ws, size_t ws_size,
                              hipStream_t stream)
{
    (void)in_sizes; (void)n_in; (void)out_size; (void)ws_size;

    const float* after    = (const float*)d_in[0];
    const float* before   = (const float*)d_in[1];
    const float* d_outs   = (const float*)d_in[2];
    /* d_in[3] = p_outs (unused by reference output) */
    const float* e_outs   = (const float*)d_in[4];
    const float* ys       = (const float*)d_in[5];
    const int*   ds       = (const int*)  d_in[6];
    /* d_in[7] = ps (unused) */
    const float* es       = (const float*)d_in[8];
    const int*   ilens    = (const int*)  d_in[9];
    const int*   olens    = (const int*)  d_in[10];
    const float* p_m_outs = (const float*)d_in[11];
    const float* pms      = (const float*)d_in[12];
    float* out = (float*)d_out;

    // Workspace layout (floats): every slot fully rewritten each launch.
    float* ws        = (float*)d_ws;
    float* fsum_ys   = ws;                            // 64*2600
    float* fsum_af   = fsum_ys + BATCH * T_FEATS;     // 64*2600
    float* l1_part   = fsum_af + BATCH * T_FEATS;     // 512
    float* dur_part  = l1_part + NBLK_A;              // 64
    float* en_part   = dur_part + BATCH;              // 64
    float* mean_part = en_part + BATCH;               // 64

    fs2_stream_kernel<<<NBLK_A, 256, 0, stream>>>(after, before, ys, olens,
                                                  fsum_ys, fsum_af, l1_part);
    fs2_token_kernel<<<BATCH, 256, 0, stream>>>(d_outs, e_outs, es, ds, ilens,
                                                fsum_ys, fsum_af,
                                                dur_part, en_part, mean_part);
    fs2_final_kernel<<<1, 32, 0, stream>>>(l1_part, dur_part, en_part, mean_part,
                                           ilens, olens, p_m_outs, pms, out);
}